// NeighborhoodPool_48687749268231
// MI455X (gfx1250) — compile-verified
//
#include <hip/hip_runtime.h>
#include <hip/hip_bf16.h>
#include <math.h>

#define N_NODES 100000
#define N_EDGES 3200000
#define D_FEAT  256

typedef __attribute__((ext_vector_type(2))) float v2f;
typedef __attribute__((ext_vector_type(8))) float v8f;

// ---- float atomic max via sign-split int/uint trick (monotone, init -inf) ----
__device__ __forceinline__ void atomicFMax(float* addr, float val) {
    if (val >= 0.0f) {
        atomicMax((int*)addr, __float_as_int(val));
    } else {
        atomicMin((unsigned int*)addr, __float_as_uint(val));
    }
}

// ---- init scratch state (re-run every launch: graph-replay safe) ----
__global__ void k_init(float* m, float* ssum, float* outv, int* cur, int* nxt,
                       float* pooled, float* gmax, float* gsum, int* argid, int n) {
    int i = blockIdx.x * blockDim.x + threadIdx.x;
    if (i < n) {
        m[i]    = -INFINITY;
        ssum[i] = 0.0f;
        outv[i] = 0.0f;
        cur[i]  = 0;
        nxt[i]  = 0;
    }
    if (i < D_FEAT) pooled[i] = -INFINITY;
    if (i == 0) { *gmax = -INFINITY; *gsum = 0.0f; *argid = 0x7fffffff; }
}

// ---- xl/xr GEMV via V_WMMA_F32_16X16X4_F32: C(16x16) += A(16x4) x B(4x16) ----
// B column 0 = w_l slice, column 1 = w_r slice, columns 2..15 zero.
// A layout: lane l holds M=l%16, K = (l/16)*2 + {0,1} in a[0],a[1]
// B layout: lane l holds N=l%16, VGPR j holds K = j + (l/16)*2
// D layout: VGPR j: lanes0-15 -> (M=j, N=lane), lanes16-31 -> (M=j+8, N=lane-16)
__global__ void k_gemv_wmma(const float* __restrict__ x,
                            const float* __restrict__ wl,
                            const float* __restrict__ wr,
                            float* __restrict__ xl, float* __restrict__ xr,
                            int nrows) {
    int wave   = (int)((blockIdx.x * blockDim.x + threadIdx.x) >> 5);
    int nwaves = nrows >> 4;                 // 100000/16 = 6250, exact
    if (wave >= nwaves) return;              // whole-wave exit: WMMA sees full EXEC
    int lane = threadIdx.x & 31;
    int mrow = lane & 15;                    // A: row within 16-row tile
    int col  = lane & 15;                    // B/D: column within tile
    int kh   = (lane >> 4) << 1;             // K sub-offset: 0 or 2

    const float* xrow = x + (size_t)(wave * 16 + mrow) * D_FEAT + kh;
    const float* wcol = (col == 0) ? wl : (col == 1) ? wr : nullptr;

    v8f acc = {};
    for (int k = 0; k < D_FEAT; k += 4) {
        v2f a; a.x = xrow[k]; a.y = xrow[k + 1];
        v2f b;
        if (wcol) { b.x = wcol[k + kh]; b.y = wcol[k + kh + 1]; }
        else      { b.x = 0.0f;         b.y = 0.0f; }
        acc = __builtin_amdgcn_wmma_f32_16x16x4_f32(
                  false, a, false, b, (short)0, acc, false, false);
    }

    int base = wave * 16 + ((lane >> 4) << 3);   // M = j (+8 for upper half-lanes)
    if (col == 0) {
#pragma unroll
        for (int j = 0; j < 8; ++j) xl[base + j] = acc[j];
    } else if (col == 1) {
#pragma unroll
        for (int j = 0; j < 8; ++j) xr[base + j] = acc[j];
    }
}

// ---- edge pass 1: edge logit + segment max ----
__global__ void k_edge_max(const int* __restrict__ src, const int* __restrict__ dst,
                           const float* __restrict__ pos, const float* __restrict__ we,
                           const float* __restrict__ att,
                           const float* __restrict__ xl, const float* __restrict__ xr,
                           float* __restrict__ ebuf, float* __restrict__ m, int ne) {
    int i = blockIdx.x * blockDim.x + threadIdx.x;
    if (i >= ne) return;
    int s = src[i], d = dst[i];
    float ea = (pos[3 * d + 0] - pos[3 * s + 0]) * we[0]
             + (pos[3 * d + 1] - pos[3 * s + 1]) * we[1]
             + (pos[3 * d + 2] - pos[3 * s + 2]) * we[2];
    float msg = xl[s] + xr[d] + ea;
    float lr  = msg > 0.0f ? msg : 0.2f * msg;   // leaky_relu, NEG_SLOPE=0.2
    float ev  = lr * att[0];
    ebuf[i] = ev;
    atomicFMax(&m[d], ev);
}

// ---- edge pass 2: exp(e - m[dst]) stored in place, segment sum ----
__global__ void k_edge_expsum(const int* __restrict__ dst,
                              float* __restrict__ ebuf, const float* __restrict__ m,
                              float* __restrict__ ssum, int ne) {
    int i = blockIdx.x * blockDim.x + threadIdx.x;
    if (i >= ne) return;
    int d = dst[i];
    float ex = expf(ebuf[i] - m[d]);
    ebuf[i] = ex;
    atomicAdd(&ssum[d], ex);
}

// ---- edge pass 3: out[dst] += alpha * xl[src] ----
__global__ void k_edge_acc(const int* __restrict__ src, const int* __restrict__ dst,
                           const float* __restrict__ ebuf, const float* __restrict__ ssum,
                           const float* __restrict__ xl, float* __restrict__ outv, int ne) {
    int i = blockIdx.x * blockDim.x + threadIdx.x;
    if (i >= ne) return;
    int s = src[i], d = dst[i];
    float alpha = ebuf[i] / (ssum[d] + 1e-16f);
    atomicAdd(&outv[d], alpha * xl[s]);
}

// ---- global softmax pass 1: max of logits ----
__global__ void k_nodemax(const float* __restrict__ outv, const float* __restrict__ bias,
                          float* gmax, int n) {
    __shared__ float red[256];
    int i = blockIdx.x * blockDim.x + threadIdx.x;
    float v = (i < n) ? (outv[i] + bias[0]) : -INFINITY;
    red[threadIdx.x] = v;
    __syncthreads();
    for (int st = 128; st > 0; st >>= 1) {
        if ((int)threadIdx.x < st)
            red[threadIdx.x] = fmaxf(red[threadIdx.x], red[threadIdx.x + st]);
        __syncthreads();
    }
    if (threadIdx.x == 0) atomicFMax(gmax, red[0]);
}

// ---- global softmax pass 2: sum of exp + first-occurrence argmax ----
__global__ void k_nodesum_argmax(const float* __restrict__ outv, const float* __restrict__ bias,
                                 const float* __restrict__ gmax, float* gsum,
                                 int* argid, int n) {
    __shared__ float red[256];
    int i = blockIdx.x * blockDim.x + threadIdx.x;
    float gm = *gmax;
    float ex = 0.0f;
    if (i < n) {
        float v = outv[i] + bias[0];
        ex = expf(v - gm);
        if (v == gm) atomicMin((unsigned int*)argid, (unsigned int)i);  // first index on ties
    }
    red[threadIdx.x] = ex;
    __syncthreads();
    for (int st = 128; st > 0; st >>= 1) {
        if ((int)threadIdx.x < st)
            red[threadIdx.x] += red[threadIdx.x + st];
        __syncthreads();
    }
    if (threadIdx.x == 0) atomicAdd(gsum, red[0]);
}

// ---- score output + seed BFS ----
__global__ void k_score(const float* __restrict__ outv, const float* __restrict__ bias,
                        const float* __restrict__ gmax, const float* __restrict__ gsum,
                        const int* __restrict__ argid, float* __restrict__ score,
                        int* __restrict__ cur, int n) {
    int i = blockIdx.x * blockDim.x + threadIdx.x;
    if (i >= n) return;
    float v = outv[i] + bias[0];
    score[i] = expf(v - *gmax) / (*gsum);
    if (i == *argid) cur[i] = 1;
}

// ---- one BFS hop: next = cur ∪ {dst : cur[src]} (next pre-copied from cur) ----
__global__ void k_bfs(const int* __restrict__ src, const int* __restrict__ dst,
                      const int* __restrict__ cur, int* __restrict__ nxt, int ne) {
    int i = blockIdx.x * blockDim.x + threadIdx.x;
    if (i >= ne) return;
    if (cur[src[i]]) nxt[dst[i]] = 1;   // same-value races are benign
}

// ---- masked feature-wise max pool; 256 threads = 256 features, coalesced rows ----
__global__ void k_pooled(const float* __restrict__ x, const int* __restrict__ reach,
                         float* __restrict__ pooled, int n) {
    int t  = threadIdx.x;                 // feature index
    int r0 = blockIdx.x * 256;
    int r1 = min(r0 + 256, n);
    float lm = -INFINITY;
    for (int r = r0; r < r1; ++r) {
        if (reach[r]) lm = fmaxf(lm, x[(size_t)r * D_FEAT + t]);
    }
    atomicFMax(&pooled[t], lm);
}

extern "C" void kernel_launch(void* const* d_in, const int* in_sizes, int n_in,
                              void* d_out, int out_size, void* d_ws, size_t ws_size,
                              hipStream_t stream) {
    const float* x    = (const float*)d_in[0];
    const float* pos  = (const float*)d_in[1];
    const float* wl   = (const float*)d_in[2];
    const float* wr   = (const float*)d_in[3];
    const float* we   = (const float*)d_in[4];
    const float* att  = (const float*)d_in[5];
    const float* bias = (const float*)d_in[6];
    const int*   ei   = (const int*)d_in[7];

    const int N = N_NODES, E = N_EDGES;
    const int* srcI = ei;
    const int* dstI = ei + E;

    float* score  = (float*)d_out;        // [N]
    float* pooled = (float*)d_out + N;    // [256]

    // workspace carve (all fp32/int32, ~15.6 MB)
    float* xl   = (float*)d_ws;           // N
    float* xr   = xl + N;                 // N
    float* m    = xr + N;                 // N
    float* ssum = m + N;                  // N
    float* outv = ssum + N;               // N
    float* ebuf = outv + N;               // E
    int*   cur  = (int*)(ebuf + E);       // N
    int*   nxt  = cur + N;                // N
    float* gmax = (float*)(nxt + N);      // 1
    float* gsum = gmax + 1;               // 1
    int*   argid = (int*)(gsum + 1);      // 1

    const int TB = 256;
    int nodeBlocks = (N + TB - 1) / TB;       // 391
    int edgeBlocks = (E + TB - 1) / TB;       // 12500
    int gemvBlocks = ((N / 16) * 32 + TB - 1) / TB;  // 6250 waves -> 782 blocks

    k_init<<<nodeBlocks, TB, 0, stream>>>(m, ssum, outv, cur, nxt, pooled,
                                          gmax, gsum, argid, N);

    k_gemv_wmma<<<gemvBlocks, TB, 0, stream>>>(x, wl, wr, xl, xr, N);

    k_edge_max<<<edgeBlocks, TB, 0, stream>>>(srcI, dstI, pos, we, att, xl, xr,
                                              ebuf, m, E);
    k_edge_expsum<<<edgeBlocks, TB, 0, stream>>>(dstI, ebuf, m, ssum, E);
    k_edge_acc<<<edgeBlocks, TB, 0, stream>>>(srcI, dstI, ebuf, ssum, xl, outv, E);

    k_nodemax<<<nodeBlocks, TB, 0, stream>>>(outv, bias, gmax, N);
    k_nodesum_argmax<<<nodeBlocks, TB, 0, stream>>>(outv, bias, gmax, gsum, argid, N);
    k_score<<<nodeBlocks, TB, 0, stream>>>(outv, bias, gmax, gsum, argid, score, cur, N);

    // exactly K_MAX = 5 hops, double-buffered to avoid over-propagation
    for (int it = 0; it < 5; ++it) {
        hipMemcpyAsync(nxt, cur, (size_t)N * sizeof(int),
                       hipMemcpyDeviceToDevice, stream);
        k_bfs<<<edgeBlocks, TB, 0, stream>>>(srcI, dstI, cur, nxt, E);
        int* t = cur; cur = nxt; nxt = t;
    }

    k_pooled<<<nodeBlocks, TB, 0, stream>>>(x, cur, pooled, N);
}